// XNMNet_27092653703937
// MI455X (gfx1250) — compile-verified
//
#include <hip/hip_runtime.h>

#define B 16
#define N 512
#define DPRE 96
#define DV 512
#define NEC 16
#define MAXREL 3
#define NATTR 64
#define NCLS 28
#define NSPLIT 8

typedef __attribute__((ext_vector_type(2))) float v2f;
typedef __attribute__((ext_vector_type(8))) float v8f;

__device__ __forceinline__ float sigmoidf(float x) {
    return 1.0f / (1.0f + __expf(-x));
}

// ---------------------------------------------------------------------------
// Kernel 1: feat = pre_v @ W_map + b_map  via V_WMMA_F32_16X16X4_F32
// pre_v: (B*N, 96) row-major; W_map: (96, 512) row-major; feat: (B*N, 512).
// Each wave computes a 16x64 output strip; K loop of 24 x 4 WMMA ops.
// A layout (ISA 7.12.2, 32-bit A 16x4): lanes 0-15 hold K={0,1} of row M=lane,
// lanes 16-31 hold K={2,3}.  B mirrors with N across lanes.
// D layout: VGPR r -> row (r + 8*(lane>=16)), col = lane&15.
// ---------------------------------------------------------------------------
__global__ __launch_bounds__(256) void k_feat_gemm(const float* __restrict__ preV,
                                                   const float* __restrict__ Wmap,
                                                   const float* __restrict__ bmap,
                                                   float* __restrict__ feat)
{
    const int lane = threadIdx.x & 31;
    const int wave = threadIdx.x >> 5;
    const int tile = blockIdx.x * 8 + wave;     // 4096 wave-tiles total
    const int rowBase = (tile >> 3) * 16;       // 512 row tiles
    const int colBase = (tile & 7) * 64;        // 8 column strips of 64
    const int m  = lane & 15;
    const int kh = lane >> 4;                   // K-half: 0 -> K{0,1}, 1 -> K{2,3}

    v8f acc0 = {}; v8f acc1 = {}; v8f acc2 = {}; v8f acc3 = {};
    const float* aRow = preV + (size_t)(rowBase + m) * DPRE + 2 * kh;

    for (int k0 = 0; k0 < DPRE; k0 += 4) {
        v2f a = *reinterpret_cast<const v2f*>(aRow + k0);   // 8B-aligned (even offset)
        const float* bp = Wmap + (size_t)(k0 + 2 * kh) * DV + colBase + m;
        v2f b0 = { bp[0],  bp[DV + 0]  };
        v2f b1 = { bp[16], bp[DV + 16] };
        v2f b2 = { bp[32], bp[DV + 32] };
        v2f b3 = { bp[48], bp[DV + 48] };
        acc0 = __builtin_amdgcn_wmma_f32_16x16x4_f32(false, a, false, b0, (short)0, acc0, false, false);
        acc1 = __builtin_amdgcn_wmma_f32_16x16x4_f32(false, a, false, b1, (short)0, acc1, false, false);
        acc2 = __builtin_amdgcn_wmma_f32_16x16x4_f32(false, a, false, b2, (short)0, acc2, false, false);
        acc3 = __builtin_amdgcn_wmma_f32_16x16x4_f32(false, a, false, b3, (short)0, acc3, false, false);
    }

    const int rowHalf = rowBase + 8 * kh;
    #pragma unroll
    for (int r = 0; r < 8; ++r) {
        float* dst = feat + (size_t)(rowHalf + r) * DV + colBase + m;
        dst[0]  = acc0[r] + bmap[colBase + m + 0];
        dst[16] = acc1[r] + bmap[colBase + m + 16];
        dst[32] = acc2[r] + bmap[colBase + m + 32];
        dst[48] = acc3[r] + bmap[colBase + m + 48];
    }
}

// ---------------------------------------------------------------------------
// Kernel 2: queries = word_emb[program_inputs]   (B*4 vectors of 512)
// ---------------------------------------------------------------------------
__global__ __launch_bounds__(128) void k_queries(const int* __restrict__ prog,
                                                 const float* __restrict__ wemb,
                                                 float* __restrict__ q)
{
    const int bq = blockIdx.x;              // b*4 + s
    const float* src = wemb + (size_t)prog[bq] * DV;
    float* dst = q + (size_t)bq * DV;
    for (int d = threadIdx.x; d < DV; d += 128) dst[d] = src[d];
}

// ---------------------------------------------------------------------------
// Kernel 3: g LUT: gtab[b][step][c] = sigmoid(ecv[c] . q[b, step==0?1:3]);
// category 0 is padding -> stored as 0 so the relate kernel needs no mask.
// ---------------------------------------------------------------------------
__global__ __launch_bounds__(128) void k_gtab(const float* __restrict__ ecv,
                                              const float* __restrict__ q,
                                              float* __restrict__ gtab)
{
    __shared__ float red[128];
    const int c = blockIdx.x;           // 0..15
    const int s = blockIdx.y;           // 0..1
    const int b = blockIdx.z;
    const int qidx = (s == 0) ? 1 : 3;
    const float* qv = q + ((size_t)b * 4 + qidx) * DV;
    const float* ev = ecv + (size_t)c * DV;
    float acc = 0.f;
    for (int d = threadIdx.x; d < DV; d += 128) acc += ev[d] * qv[d];
    red[threadIdx.x] = acc; __syncthreads();
    for (int st = 64; st > 0; st >>= 1) {
        if (threadIdx.x < st) red[threadIdx.x] += red[threadIdx.x + st];
        __syncthreads();
    }
    if (threadIdx.x == 0)
        gtab[((size_t)b * 2 + s) * NEC + c] = (c == 0) ? 0.f : sigmoidf(red[0]);
}

// ---------------------------------------------------------------------------
// Kernel 4: filter scores.  attn1 = node_mask * sigmoid(feat.q0); f2 = sigmoid(feat.q2)
// ---------------------------------------------------------------------------
__global__ __launch_bounds__(128) void k_filter(const float* __restrict__ feat,
                                                const float* __restrict__ q,
                                                float* __restrict__ attn1,
                                                float* __restrict__ f2)
{
    __shared__ float r0[128], r2[128];
    const int n = blockIdx.x & (N - 1);
    const int b = blockIdx.x >> 9;
    const float* fv = feat + ((size_t)b * N + n) * DV;
    const float* q0 = q + ((size_t)b * 4 + 0) * DV;
    const float* q2 = q + ((size_t)b * 4 + 2) * DV;
    float a0 = 0.f, a2 = 0.f;
    for (int d = threadIdx.x; d < DV; d += 128) {
        float f = fv[d];
        a0 += f * q0[d];
        a2 += f * q2[d];
    }
    r0[threadIdx.x] = a0; r2[threadIdx.x] = a2; __syncthreads();
    for (int st = 64; st > 0; st >>= 1) {
        if (threadIdx.x < st) {
            r0[threadIdx.x] += r0[threadIdx.x + st];
            r2[threadIdx.x] += r2[threadIdx.x + st];
        }
        __syncthreads();
    }
    if (threadIdx.x == 0) {
        attn1[(size_t)b * N + n] = (n < N - NATTR) ? sigmoidf(r0[0]) : 0.f;
        f2[(size_t)b * N + n]    = sigmoidf(r2[0]);
    }
}

// ---------------------------------------------------------------------------
// Kernel 5: fused relate.  partial[b][ns][m] = sum_{n in split} attn[b,n] *
//   conn[b,n,m] * (g[cat0]+g[cat1]+g[cat2]).  Streams cat/conn exactly once.
//   Skips n-rows with attn==0 (uniform branch) and cat loads where conn==0.
// ---------------------------------------------------------------------------
__global__ __launch_bounds__(256) void k_relate(const float* __restrict__ attn,
                                                const float* __restrict__ gtab, int step,
                                                const int* __restrict__ cat,
                                                const float* __restrict__ conn,
                                                float* __restrict__ partial)
{
    __shared__ float gs[NEC];
    __shared__ float as[N / NSPLIT];
    const int m  = blockIdx.x * 256 + threadIdx.x;
    const int ns = blockIdx.y;
    const int b  = blockIdx.z;
    const int n0 = ns * (N / NSPLIT);
    if (threadIdx.x < NEC)
        gs[threadIdx.x] = gtab[((size_t)b * 2 + step) * NEC + threadIdx.x];
    if (threadIdx.x < N / NSPLIT)
        as[threadIdx.x] = attn[(size_t)b * N + n0 + threadIdx.x];
    __syncthreads();

    float acc = 0.f;
    for (int i = 0; i < N / NSPLIT; ++i) {
        const float a = as[i];              // uniform across the block
        if (a == 0.f) continue;
        const size_t base = ((size_t)b * N + (n0 + i)) * N + m;
        const float w = conn[base];
        if (w != 0.f) {
            const int* cp = cat + base * MAXREL;
            acc += a * w * (gs[cp[0]] + gs[cp[1]] + gs[cp[2]]);
        }
    }
    partial[((size_t)b * NSPLIT + ns) * N + m] = acc;
}

// ---------------------------------------------------------------------------
// Kernel 6: sum partials, normalize by row max (+1e-6), optional filter mult.
// ---------------------------------------------------------------------------
__global__ __launch_bounds__(512) void k_norm(const float* __restrict__ partial,
                                              const float* __restrict__ filt,
                                              float* __restrict__ attn_out)
{
    __shared__ float red[512];
    const int b = blockIdx.x;
    const int m = threadIdx.x;
    float s = 0.f;
    for (int i = 0; i < NSPLIT; ++i)
        s += partial[((size_t)b * NSPLIT + i) * N + m];
    red[m] = s; __syncthreads();
    for (int st = 256; st > 0; st >>= 1) {
        if (m < st) red[m] = fmaxf(red[m], red[m + st]);
        __syncthreads();
    }
    float v = s / (red[0] + 1e-6f);
    if (filt) v *= filt[(size_t)b * N + m];
    attn_out[(size_t)b * N + m] = v;
}

// ---------------------------------------------------------------------------
// Kernel 7: out[b][d] = sum_n attn[b][n] * feat[b][n][d]
// ---------------------------------------------------------------------------
__global__ __launch_bounds__(512) void k_readout(const float* __restrict__ attn,
                                                 const float* __restrict__ feat,
                                                 float* __restrict__ outbd)
{
    __shared__ float as[N];
    const int b = blockIdx.x;
    const int d = threadIdx.x;
    as[d] = attn[(size_t)b * N + d];     // N == DV == 512
    __syncthreads();
    float acc = 0.f;
    for (int n = 0; n < N; ++n)
        acc += as[n] * feat[((size_t)b * N + n) * DV + d];
    outbd[(size_t)b * DV + d] = acc;
}

// ---------------------------------------------------------------------------
// Kernel 8: classifier: relu(out @ W1 + b1) @ W2 + b2   -> (B, 28)
// ---------------------------------------------------------------------------
__global__ __launch_bounds__(256) void k_classify(const float* __restrict__ outbd,
                                                  const float* __restrict__ W1,
                                                  const float* __restrict__ b1,
                                                  const float* __restrict__ W2,
                                                  const float* __restrict__ b2,
                                                  float* __restrict__ out)
{
    __shared__ float xs[DV];
    __shared__ float hs[256];
    const int b = blockIdx.x;
    for (int d = threadIdx.x; d < DV; d += 256) xs[d] = outbd[(size_t)b * DV + d];
    __syncthreads();
    const int j = threadIdx.x;
    float h = b1[j];
    for (int d = 0; d < DV; ++d) h += xs[d] * W1[(size_t)d * 256 + j];
    hs[j] = fmaxf(h, 0.f);
    __syncthreads();
    if (j < NCLS) {
        float acc = b2[j];
        for (int k = 0; k < 256; ++k) acc += hs[k] * W2[(size_t)k * NCLS + j];
        out[(size_t)b * NCLS + j] = acc;
    }
}

// ---------------------------------------------------------------------------
extern "C" void kernel_launch(void* const* d_in, const int* in_sizes, int n_in,
                              void* d_out, int out_size, void* d_ws, size_t ws_size,
                              hipStream_t stream)
{
    const float* pre_v = (const float*)d_in[0];
    const float* conn  = (const float*)d_in[1];
    const int*   cat   = (const int*)  d_in[2];
    const int*   prog  = (const int*)  d_in[3];
    const float* wemb  = (const float*)d_in[4];
    const float* ecv   = (const float*)d_in[5];
    const float* Wmap  = (const float*)d_in[6];
    const float* bmap  = (const float*)d_in[7];
    const float* W1    = (const float*)d_in[8];
    const float* b1    = (const float*)d_in[9];
    const float* W2    = (const float*)d_in[10];
    const float* b2    = (const float*)d_in[11];
    float* out = (float*)d_out;

    // Workspace carve-up (floats), ~17.3 MB total
    float* ws      = (float*)d_ws;
    float* feat    = ws;                              // B*N*DV
    float* q       = feat    + (size_t)B * N * DV;    // B*4*DV
    float* gtab    = q       + (size_t)B * 4 * DV;    // B*2*NEC
    float* f2      = gtab    + (size_t)B * 2 * NEC;   // B*N
    float* attn1   = f2      + (size_t)B * N;         // B*N
    float* partial = attn1   + (size_t)B * N;         // B*NSPLIT*N
    float* attn3   = partial + (size_t)B * NSPLIT * N;// B*N
    float* attn4   = attn3   + (size_t)B * N;         // B*N
    float* outbd   = attn4   + (size_t)B * N;         // B*DV

    k_feat_gemm<<<512, 256, 0, stream>>>(pre_v, Wmap, bmap, feat);
    k_queries <<<B * 4, 128, 0, stream>>>(prog, wemb, q);
    k_gtab    <<<dim3(NEC, 2, B), 128, 0, stream>>>(ecv, q, gtab);
    k_filter  <<<B * N, 128, 0, stream>>>(feat, q, attn1, f2);
    k_relate  <<<dim3(N / 256, NSPLIT, B), 256, 0, stream>>>(attn1, gtab, 0, cat, conn, partial);
    k_norm    <<<B, 512, 0, stream>>>(partial, f2, attn3);
    k_relate  <<<dim3(N / 256, NSPLIT, B), 256, 0, stream>>>(attn3, gtab, 1, cat, conn, partial);
    k_norm    <<<B, 512, 0, stream>>>(partial, nullptr, attn4);
    k_readout <<<B, 512, 0, stream>>>(attn4, feat, outbd);
    k_classify<<<B, 256, 0, stream>>>(outbd, W1, b1, W2, b2, out);
}